// CNNTeacherModel_62319975465511
// MI455X (gfx1250) — compile-verified
//
#include <hip/hip_runtime.h>
#include <hip/hip_bf16.h>
#include <math.h>

typedef __attribute__((ext_vector_type(2))) float v2f;
typedef __attribute__((ext_vector_type(4))) float f4;
typedef __attribute__((ext_vector_type(8))) float v8f;

#define B_  8192
#define H_  1024
#define N_  4096
#define G_  5
#define KC  512            // K-chunk staged in LDS (32 KB)
#define CPB 128            // columns per block (8 waves x 16)
#define MAXT 544           // >= floor(B/16)+G = 517 row tiles

// ws layout (ints): [0] numTiles | [1..MAXT] tileGroup | [1+MAXT..2MAXT] tileBase
//                   [1+2*MAXT ...] rowPerm (B_ + 16*G_ padded entries)

__global__ __launch_bounds__(256)
void bucket_kernel(const int* __restrict__ groups, int* __restrict__ ws,
                   float* __restrict__ loss_out)
{
    __shared__ int cnt[G_], base[G_], cur[G_], totpad;
    const int t = threadIdx.x;
    if (t < G_) { cnt[t] = 0; cur[t] = 0; }
    __syncthreads();
    for (int i = t; i < B_; i += blockDim.x) atomicAdd(&cnt[groups[i]], 1);
    __syncthreads();
    int* tileGroup = ws + 1;
    int* tileBase  = ws + 1 + MAXT;
    int* rowPerm   = ws + 1 + 2 * MAXT;
    if (t == 0) {
        int off = 0, tt = 0;
        for (int g = 0; g < G_; ++g) {
            base[g] = off;
            int pl = ((cnt[g] + 15) >> 4) << 4;           // pad to 16
            for (int k = 0; k < pl; k += 16) { tileGroup[tt] = g; tileBase[tt] = off + k; ++tt; }
            off += pl;
        }
        ws[0]  = tt;
        totpad = off;
        *loss_out = 0.0f;                                  // init loss accumulator
    }
    __syncthreads();
    const int tot = totpad;
    for (int i = t; i < tot; i += blockDim.x) rowPerm[i] = -1;
    __syncthreads();
    for (int i = t; i < B_; i += blockDim.x) {
        int g = groups[i];
        int p = atomicAdd(&cur[g], 1);
        rowPerm[base[g] + p] = i;
    }
}

__global__ __launch_bounds__(256)
void grouped_gemm_kernel(const float* __restrict__ hidden,
                         const float* __restrict__ W,
                         const float* __restrict__ bias,
                         const int* __restrict__ ws,
                         float* __restrict__ out)
{
    const int tileIdx  = blockIdx.y;
    const int numTiles = ws[0];
    if (tileIdx >= numTiles) return;                       // uniform per block

    const int* tileGroup = ws + 1;
    const int* tileBase  = ws + 1 + MAXT;
    const int* rowPerm   = ws + 1 + 2 * MAXT;
    const int g     = tileGroup[tileIdx];
    const int rbase = tileBase[tileIdx];

    // A staged float4-blocked + XOR swizzle: block q = k4*16 + (m ^ (k4&15)),
    // byte addr = k4*256 + 16*(m ^ (k4&15)). Conflict-free for both the b128
    // cooperative writes and the b64 WMMA-operand reads.
    __shared__ float Alds[KC * 16];

    const int t    = threadIdx.x;
    const int lane = t & 31;
    const int wv   = t >> 5;               // wave 0..7
    const int half = lane >> 4;            // 0: K pair {0,1}, 1: K pair {2,3}
    const int l16  = lane & 15;
    const int col  = blockIdx.x * CPB + wv * 16 + l16;

    // Per-lane swizzled LDS byte term, period 16 in k4: precompute once.
    int laddr[16];
    #pragma unroll
    for (int p = 0; p < 16; ++p) laddr[p] = ((l16 ^ p) << 4) + 8 * half;

    const float* __restrict__ Wg = W + (size_t)g * H_ * N_;
    v8f acc = {};

    for (int ks = 0; ks < H_; ks += KC) {
        __syncthreads();                   // LDS reuse fence
        // 16 rows x KC floats = 2048 float4; 256 threads -> 8 each, coalesced per row
        #pragma unroll
        for (int j = 0; j < (16 * KC / 4) / 256; ++j) {
            int flat = j * 256 + t;        // 0..2047
            int m    = flat >> 7;          // KC/4 = 128 float4 per row
            int k4   = flat & 127;
            int row  = rowPerm[rbase + m];
            f4 v = {0.f, 0.f, 0.f, 0.f};
            if (row >= 0)
                v = *(const f4*)(hidden + (size_t)row * H_ + ks + k4 * 4);
            int q = k4 * 16 + (m ^ (k4 & 15));
            *(f4*)(&Alds[q * 4]) = v;
        }
        __syncthreads();

        // Per-lane B base for this stage: rows ks+2*half / +1, column `col`.
        const float* __restrict__ Bbase = Wg + (size_t)(ks + 2 * half) * N_ + col;
        const char*  __restrict__ Ab    = (const char*)Alds;

        for (int kb = 0; kb < KC; kb += 64) {              // 8 iterations
            const char*  Abb = Ab + kb * 64;               // k4 block base (bytes)
            const float* Bbb = Bbase + (size_t)kb * N_;
            #pragma unroll
            for (int p = 0; p < 16; ++p) {                 // k4 = kb/4 + p
                v2f a = *(const v2f*)(Abb + p * 256 + laddr[p]);
                v2f b;
                b.x = Bbb[(size_t)(4 * p) * N_];           // imm offset p*65536
                b.y = Bbb[(size_t)(4 * p + 1) * N_];       // imm offset +16384
                acc = __builtin_amdgcn_wmma_f32_16x16x4_f32(
                    false, a, false, b, (short)0, acc, false, false);
            }
        }
    }

    const float bv = bias[g * N_ + col];
    #pragma unroll
    for (int r = 0; r < 8; ++r) acc[r] += bv;

    // C/D layout: VGPR r holds M=r (lanes 0-15) and M=r+8 (lanes 16-31), N=l16
    #pragma unroll
    for (int r = 0; r < 8; ++r) {
        int m   = r + 8 * half;
        int row = rowPerm[rbase + m];
        if (row >= 0)
            out[(size_t)row * N_ + col] = acc[r];
    }
}

__global__ __launch_bounds__(256)
void loss_kernel(const float* __restrict__ logits,
                 const int* __restrict__ labels,
                 float* __restrict__ loss)
{
    const int b = blockIdx.x;
    const float* row = logits + (size_t)b * N_;
    __shared__ float red[256];
    const int t = threadIdx.x;

    float mx = -INFINITY;
    for (int i = t; i < N_; i += 256) mx = fmaxf(mx, row[i]);
    red[t] = mx; __syncthreads();
    for (int s = 128; s > 0; s >>= 1) {
        if (t < s) red[t] = fmaxf(red[t], red[t + s]);
        __syncthreads();
    }
    mx = red[0]; __syncthreads();

    float sum = 0.f;
    for (int i = t; i < N_; i += 256) sum += __expf(row[i] - mx);
    red[t] = sum; __syncthreads();
    for (int s = 128; s > 0; s >>= 1) {
        if (t < s) red[t] += red[t + s];
        __syncthreads();
    }
    if (t == 0) {
        float lse = mx + __logf(red[0]);
        float nll = lse - row[labels[b]];
        atomicAdd(loss, nll * (1.0f / B_));
    }
}

extern "C" void kernel_launch(void* const* d_in, const int* in_sizes, int n_in,
                              void* d_out, int out_size, void* d_ws, size_t ws_size,
                              hipStream_t stream)
{
    const float* hidden = (const float*)d_in[0];   // [B,H]
    const float* W      = (const float*)d_in[1];   // [G,H,N]
    const float* bias   = (const float*)d_in[2];   // [G,N]
    const int*   groups = (const int*)d_in[3];     // [B]
    const int*   labels = (const int*)d_in[4];     // [B]
    float* out  = (float*)d_out;                   // [B*N] logits ++ [1] loss
    float* loss = out + (size_t)B_ * N_;
    int*   ws   = (int*)d_ws;

    bucket_kernel<<<1, 256, 0, stream>>>(groups, ws, loss);

    dim3 grid(N_ / CPB, MAXT);
    grouped_gemm_kernel<<<grid, 256, 0, stream>>>(hidden, W, bias, ws, out);

    loss_kernel<<<B_, 256, 0, stream>>>(out, labels, loss);
}